// real_KW8_52699248721963
// MI455X (gfx1250) — compile-verified
//
#include <hip/hip_runtime.h>
#include <math.h>

// ---------------- configuration ----------------
#define NT      320            // threads per block: 10 wave32s; 32000 = 320*4*25
#define HEADS   8
#define INV_T   0.05f          // 1/T, T = 20
#define NEG_BIG (-3.0e38f)     // identity max (finite, so exp() never sees NaN)

// LDS layout (floats), dynamic shared:
//   [0 .. BUF-1]            pass-2 double buffers: 2 * 8 heads * CHUNK floats
//                           (pass-1 aliases this region: mArr[9*512], sArr[9*512])
//   [SUM_OFF .. +511]       final sum reduction tree
//   [FIN_OFF .. +31]        broadcast finals: fMax[9] at +0, fInvZ[9] at +16
#define CHUNK       (NT * 4)                 // 1280 floats per head per chunk
#define BUF_FLOATS  (2 * HEADS * CHUNK)      // 20480
#define SUM_OFF     BUF_FLOATS
#define FIN_OFF     (BUF_FLOATS + 512)
#define SMEM_FLOATS (BUF_FLOATS + 512 + 32)  // 21024 floats = 84096 bytes

// Async global->LDS path (CDNA5): guarded so either toolchain compiles.
#if defined(__HIP_DEVICE_COMPILE__) && defined(__gfx1250__) && \
    __has_builtin(__builtin_amdgcn_global_load_async_to_lds_b128) && \
    __has_builtin(__builtin_amdgcn_s_wait_asynccnt)
#define HAVE_ASYNC_LDS 1
typedef int v4i __attribute__((ext_vector_type(4)));      // builtin pointee type (global arg)
typedef __attribute__((address_space(3))) v4i lv4i;       // builtin pointee type (LDS arg)
#else
#define HAVE_ASYNC_LDS 0
#endif

__device__ __forceinline__ float elem4(const float4& v, int e) {
  switch (e & 3) { case 0: return v.x; case 1: return v.y; case 2: return v.z; default: return v.w; }
}

// Online softmax accumulate: running (max, sum of exp((x-max)/T)).
__device__ __forceinline__ void online_update(float& mx, float& s, float x) {
  float nm = fmaxf(mx, x);
  s = s * __expf((mx - nm) * INV_T) + __expf((x - nm) * INV_T);
  mx = nm;
}

// Merge two online-softmax states.
__device__ __forceinline__ void merge_state(float& m1, float& s1, float m2, float s2) {
  float M = fmaxf(m1, m2);
  s1 = s1 * __expf((m1 - M) * INV_T) + s2 * __expf((m2 - M) * INV_T);
  m1 = M;
}

// Per-float4-column-group contribution to sum_h sum_c |p_h - p_m|.
__device__ __forceinline__ float l1_contrib(const float4* xs, const float* fm, const float* fz) {
  float a = 0.f;
#pragma unroll
  for (int e = 0; e < 4; ++e) {
    float m = 0.f;
    float px[HEADS];
#pragma unroll
    for (int h = 0; h < HEADS; ++h) {
      float x = elem4(xs[h], e);
      m += x;
      px[h] = __expf((x - fm[h]) * INV_T) * fz[h];
    }
    float pm = __expf((m * 0.125f - fm[8]) * INV_T) * fz[8];
#pragma unroll
    for (int h = 0; h < HEADS; ++h) a += fabsf(px[h] - pm);
  }
  return a;
}

__global__ __launch_bounds__(NT) void kw8_kernel(
    const float* __restrict__ i0, const float* __restrict__ i1,
    const float* __restrict__ i2, const float* __restrict__ i3,
    const float* __restrict__ i4, const float* __restrict__ i5,
    const float* __restrict__ i6, const float* __restrict__ i7,
    float* __restrict__ out, int C)
{
  extern __shared__ float smem[];
  const int tid = threadIdx.x;
  const int row = blockIdx.x;
  const long rowBase = (long)row * (long)C;

  const float* in[HEADS] = { i0 + rowBase, i1 + rowBase, i2 + rowBase, i3 + rowBase,
                             i4 + rowBase, i5 + rowBase, i6 + rowBase, i7 + rowBase };

  // ---------------- Pass 1: fused online max + Z for 9 streams (8 heads + mimic) ----
  float mx[9], ss[9];
#pragma unroll
  for (int j = 0; j < 9; ++j) { mx[j] = NEG_BIG; ss[j] = 0.f; }

  for (int base = tid * 4; base + 3 < C; base += NT * 4) {
    float4 xs[HEADS];
#pragma unroll
    for (int h = 0; h < HEADS; ++h) {
      xs[h] = *reinterpret_cast<const float4*>(in[h] + base);
      __builtin_prefetch(in[h] + base + NT * 4, 0, 0);   // -> global_prefetch
    }
#pragma unroll
    for (int e = 0; e < 4; ++e) {
      float m = 0.f;
#pragma unroll
      for (int h = 0; h < HEADS; ++h) {
        float x = elem4(xs[h], e);
        m += x;
        online_update(mx[h], ss[h], x);
      }
      online_update(mx[8], ss[8], m * 0.125f);
    }
  }

  // Cross-thread merge via padded-512 LDS tree.
  float* mArr = smem;              // 9*512 floats
  float* sArr = smem + 9 * 512;    // 9*512 floats (all inside the 20480-float buffer region)
#pragma unroll
  for (int j = 0; j < 9; ++j) {
    mArr[j * 512 + tid] = mx[j];
    sArr[j * 512 + tid] = ss[j];
  }
  for (int pad = tid + NT; pad < 512; pad += NT) {
#pragma unroll
    for (int j = 0; j < 9; ++j) { mArr[j * 512 + pad] = NEG_BIG; sArr[j * 512 + pad] = 0.f; }
  }
  __syncthreads();
  for (int off = 256; off > 0; off >>= 1) {
    if (tid < off) {
#pragma unroll
      for (int j = 0; j < 9; ++j) {
        float m1 = mArr[j * 512 + tid], s1 = sArr[j * 512 + tid];
        merge_state(m1, s1, mArr[j * 512 + tid + off], sArr[j * 512 + tid + off]);
        mArr[j * 512 + tid] = m1; sArr[j * 512 + tid] = s1;
      }
    }
    __syncthreads();
  }
  if (tid < 9) {
    smem[FIN_OFF + tid]      = mArr[tid * 512];          // fMax[j]
    smem[FIN_OFF + 16 + tid] = 1.0f / sArr[tid * 512];   // fInvZ[j]
  }
  __syncthreads();

  float fm[9], fz[9];
#pragma unroll
  for (int j = 0; j < 9; ++j) { fm[j] = smem[FIN_OFF + j]; fz[j] = smem[FIN_OFF + 16 + j]; }

  // ---------------- Pass 2: L1 distance to mimic distribution -----------------------
  float acc = 0.f;

#if HAVE_ASYNC_LDS
  if ((C % CHUNK) == 0) {
    const int nIter = C / CHUNK;
    // Prologue: async-copy chunk 0 into buffer 0 (each lane moves 16B per head).
#pragma unroll
    for (int h = 0; h < HEADS; ++h)
      __builtin_amdgcn_global_load_async_to_lds_b128(
          (v4i*)(in[h] + tid * 4),
          (lv4i*)&smem[h * CHUNK + tid * 4], 0, 0);
    int sel = 0;
    for (int it = 0; it < nIter; ++it) {
      if (it + 1 < nIter) {
        const int nbase = (it + 1) * CHUNK + tid * 4;
        const int nsel = sel ^ 1;
#pragma unroll
        for (int h = 0; h < HEADS; ++h)
          __builtin_amdgcn_global_load_async_to_lds_b128(
              (v4i*)(in[h] + nbase),
              (lv4i*)&smem[nsel * HEADS * CHUNK + h * CHUNK + tid * 4], 0, 0);
        __builtin_amdgcn_s_wait_asynccnt(8);   // current chunk's 8 copies done (in-order)
      } else {
        __builtin_amdgcn_s_wait_asynccnt(0);
      }
      __syncthreads();                          // all waves' copies visible
      float4 xs[HEADS];
#pragma unroll
      for (int h = 0; h < HEADS; ++h)
        xs[h] = *reinterpret_cast<const float4*>(&smem[sel * HEADS * CHUNK + h * CHUNK + tid * 4]);
      acc += l1_contrib(xs, fm, fz);
      __syncthreads();                          // done reading before buffer is reused
      sel ^= 1;
    }
  } else
#endif
  {
    for (int base = tid * 4; base + 3 < C; base += NT * 4) {
      float4 xs[HEADS];
#pragma unroll
      for (int h = 0; h < HEADS; ++h)
        xs[h] = *reinterpret_cast<const float4*>(in[h] + base);
      acc += l1_contrib(xs, fm, fz);
    }
  }

  // Final sum reduction across the workgroup.
  float* sumArr = smem + SUM_OFF;
  sumArr[tid] = acc;
  for (int pad = tid + NT; pad < 512; pad += NT) sumArr[pad] = 0.f;
  __syncthreads();
  for (int off = 256; off > 0; off >>= 1) {
    if (tid < off) sumArr[tid] += sumArr[tid + off];
    __syncthreads();
  }
  if (tid == 0) out[row] = sumArr[0] * 0.125f;   // mean over 8 heads
}

extern "C" void kernel_launch(void* const* d_in, const int* in_sizes, int n_in,
                              void* d_out, int out_size, void* d_ws, size_t ws_size,
                              hipStream_t stream) {
  (void)n_in; (void)d_ws; (void)ws_size;
  const float* i0 = (const float*)d_in[0];
  const float* i1 = (const float*)d_in[1];
  const float* i2 = (const float*)d_in[2];
  const float* i3 = (const float*)d_in[3];
  const float* i4 = (const float*)d_in[4];
  const float* i5 = (const float*)d_in[5];
  const float* i6 = (const float*)d_in[6];
  const float* i7 = (const float*)d_in[7];
  float* out = (float*)d_out;

  const int B = out_size;                 // 1024 rows (real_KW is empty)
  const int C = in_sizes[0] / B;          // 32000 columns

  kw8_kernel<<<dim3(B), dim3(NT), SMEM_FLOATS * sizeof(float), stream>>>(
      i0, i1, i2, i3, i4, i5, i6, i7, out, C);
}